// GCNContrastive_65687229826124
// MI455X (gfx1250) — compile-verified
//
#include <hip/hip_runtime.h>

typedef float v2f __attribute__((ext_vector_type(2)));
typedef float v8f __attribute__((ext_vector_type(8)));

#define N_NODES 50000
#define N_EDGES 800000
#define IN_F    256
#define HID_F   128
#define OUT_F   64

// ---------------- utility kernels ----------------

__global__ void fill_zero(float* __restrict__ p, int n) {
  int i = blockIdx.x * blockDim.x + threadIdx.x;
  if (i < n) p[i] = 0.0f;
}

__global__ void degree_kernel(const int* __restrict__ src, const int* __restrict__ dst,
                              float* __restrict__ deg_out, float* __restrict__ deg_in, int E) {
  int e = blockIdx.x * blockDim.x + threadIdx.x;
  if (e < E) {
    atomicAdd(&deg_out[src[e]], 1.0f);
    atomicAdd(&deg_in[dst[e]], 1.0f);
  }
}

__global__ void rinv_kernel(float* __restrict__ deg, int n) {
  int i = blockIdx.x * blockDim.x + threadIdx.x;
  if (i < n) deg[i] = rsqrtf(fmaxf(deg[i], 1.0f));
}

// ---------------- fp32 WMMA GEMM: C[M,N] = (A[M,K] @ B[K,N]) * rowscale[m] ----------------
// One 16x16 output tile per wave; blockDim.x/32 waves cover N/16 tiles; blockIdx.x = M-tile.
// M must be a multiple of 16 (50000 = 3125*16), K a multiple of 4 -> EXEC stays all-ones.
__global__ void gemm_wmma_rowscale(const float* __restrict__ A, const float* __restrict__ B,
                                   const float* __restrict__ rowscale, float* __restrict__ C,
                                   int N, int K) {
  const int lane   = threadIdx.x & 31;
  const int wave   = threadIdx.x >> 5;
  const int lane16 = lane & 15;
  const int half   = lane >> 4;          // 0: lanes 0-15, 1: lanes 16-31
  const int tm = blockIdx.x;             // M tile
  const int tn = wave;                   // N tile
  const int rowA = tm * 16 + lane16;     // A fragment: M = lane%16
  const int colB = tn * 16 + lane16;     // B fragment: N = lane%16
  const float* Arow = A + (size_t)rowA * K;

  v8f acc = {};
  for (int k0 = 0; k0 < K; k0 += 4) {
    const int ka = k0 + half * 2;        // lanes 0-15: K0,K1 ; lanes 16-31: K2,K3
    v2f a = *(const v2f*)(Arow + ka);    // 8B-aligned (ka even, K even)
    v2f b;
    b.x = B[(size_t)(ka + 0) * N + colB];
    b.y = B[(size_t)(ka + 1) * N + colB];
    acc = __builtin_amdgcn_wmma_f32_16x16x4_f32(false, a, false, b, (short)0, acc,
                                                false, false);
  }
  // C/D layout: VGPR j -> M = j (lanes 0-15) / j+8 (lanes 16-31), N = lane%16
#pragma unroll
  for (int j = 0; j < 8; ++j) {
    const int m = tm * 16 + j + half * 8;
    C[(size_t)m * N + tn * 16 + lane16] = acc[j] * rowscale[m];
  }
}

// ---------------- edge-parallel scatter-add: agg[dst] += h[src] ----------------
// Thread handles one float4 feature group of one edge; a wave covers contiguous
// feature groups of the same edge -> coalesced gather, f32 atomic scatter.
__global__ void scatter_add(const float* __restrict__ h, const int* __restrict__ src,
                            const int* __restrict__ dst, float* __restrict__ agg,
                            int E, int F, int gshift) {
  const int groups = F >> 2;
  const int tid = blockIdx.x * blockDim.x + threadIdx.x;
  const int e = tid >> gshift;
  const int g = tid & (groups - 1);
  if (e >= E) return;
  const int s = src[e];
  const int d = dst[e];
  const float4 v = *(const float4*)(h + (size_t)s * F + (g << 2));
  float* p = agg + (size_t)d * F + (g << 2);
  atomicAdd(p + 0, v.x);
  atomicAdd(p + 1, v.y);
  atomicAdd(p + 2, v.z);
  atomicAdd(p + 3, v.w);
}

// ---------------- fused epilogues ----------------

__global__ void epilogue_relu(float* __restrict__ buf, const float* __restrict__ rinv_in,
                              const float* __restrict__ bias, int n, int fshift, int fmask) {
  int i = blockIdx.x * blockDim.x + threadIdx.x;
  if (i < n) {
    float v = buf[i] * rinv_in[i >> fshift] + bias[i & fmask];
    buf[i] = fmaxf(v, 0.0f);
  }
}

__global__ void epilogue_hidden(const float* __restrict__ agg, const float* __restrict__ rinv_in,
                                const float* __restrict__ bias, float* __restrict__ hidden,
                                int n) {
  int i = blockIdx.x * blockDim.x + threadIdx.x;
  if (i < n) hidden[i] = agg[i] * rinv_in[i >> 6] + bias[i & 63];
}

__global__ void logits_kernel(const float* __restrict__ hidden, const float* __restrict__ Wf,
                              const float* __restrict__ bf, float* __restrict__ logits, int n) {
  int i = blockIdx.x * blockDim.x + threadIdx.x;
  if (i < n) {
    const float* h = hidden + (size_t)i * OUT_F;
    float a0 = bf[0], a1 = bf[1];
#pragma unroll 8
    for (int c = 0; c < OUT_F; ++c) {
      float hv = h[c];
      a0 += hv * Wf[c * 2 + 0];
      a1 += hv * Wf[c * 2 + 1];
    }
    logits[i * 2 + 0] = a0;
    logits[i * 2 + 1] = a1;
  }
}

// ---------------- launch ----------------

extern "C" void kernel_launch(void* const* d_in, const int* in_sizes, int n_in,
                              void* d_out, int out_size, void* d_ws, size_t ws_size,
                              hipStream_t stream) {
  (void)in_sizes; (void)n_in; (void)out_size; (void)ws_size;
  const float* x  = (const float*)d_in[0];
  const float* W1 = (const float*)d_in[1];
  const float* b1 = (const float*)d_in[2];
  const float* W2 = (const float*)d_in[3];
  const float* b2 = (const float*)d_in[4];
  const float* Wf = (const float*)d_in[5];
  const float* bf = (const float*)d_in[6];
  const int*  src = (const int*)d_in[7];
  const int*  dst = (const int*)d_in[8];

  float* out = (float*)d_out;
  float* logits = out;                          // [50000, 2]
  float* hidden = out + (size_t)N_NODES * 2;    // [50000, 64]

  float* ws       = (float*)d_ws;
  float* rinv_out = ws;                                   // 50000
  float* rinv_in  = ws + N_NODES;                         // 50000
  float* bufA     = ws + 2 * N_NODES;                     // 50000*128 (h, then h2+agg2)
  float* bufB     = bufA + (size_t)N_NODES * HID_F;       // 50000*128 (agg1 -> h1)
  float* h2       = bufA;                                 // 50000*64 (h dead by then)
  float* agg2     = bufA + (size_t)N_NODES * OUT_F;       // 50000*64

  // 1) zero degrees + bufA (must re-zero every call: atomics accumulate)
  {
    int zn = 2 * N_NODES + N_NODES * HID_F;
    fill_zero<<<(zn + 255) / 256, 256, 0, stream>>>(ws, zn);
  }
  // 2) degree counts via f32 atomics
  degree_kernel<<<(N_EDGES + 255) / 256, 256, 0, stream>>>(src, dst, rinv_out, rinv_in, N_EDGES);
  // 3) rsqrt(clip(deg,1)) in place (both arrays are contiguous)
  rinv_kernel<<<(2 * N_NODES + 255) / 256, 256, 0, stream>>>(ws, 2 * N_NODES);
  // 4) GEMM1: bufA = (x @ W1) * rsqrt(deg_out)   [50000x256 x 256x128]
  gemm_wmma_rowscale<<<N_NODES / 16, 32 * (HID_F / 16), 0, stream>>>(x, W1, rinv_out, bufA,
                                                                     HID_F, IN_F);
  // 5) aggregate layer 1: bufB[dst] += bufA[src]  (128 feats, 32 float4 groups/edge)
  {
    long long total = (long long)N_EDGES * (HID_F / 4);
    scatter_add<<<(int)((total + 255) / 256), 256, 0, stream>>>(bufA, src, dst, bufB,
                                                                N_EDGES, HID_F, 5);
  }
  // 6) h1 = relu(agg1 * rsqrt(deg_in) + b1), in place; then zero agg2
  {
    int n1 = N_NODES * HID_F;
    epilogue_relu<<<(n1 + 255) / 256, 256, 0, stream>>>(bufB, rinv_in, b1, n1, 7, 127);
    int n2 = N_NODES * OUT_F;
    fill_zero<<<(n2 + 255) / 256, 256, 0, stream>>>(agg2, n2);
  }
  // 7) GEMM2: h2 = (h1 @ W2) * rsqrt(deg_out)   [50000x128 x 128x64]
  gemm_wmma_rowscale<<<N_NODES / 16, 32 * (OUT_F / 16), 0, stream>>>(bufB, W2, rinv_out, h2,
                                                                     OUT_F, HID_F);
  // 8) aggregate layer 2: agg2[dst] += h2[src]  (64 feats, 16 float4 groups/edge)
  {
    long long total = (long long)N_EDGES * (OUT_F / 4);
    scatter_add<<<(int)((total + 255) / 256), 256, 0, stream>>>(h2, src, dst, agg2,
                                                                N_EDGES, OUT_F, 4);
  }
  // 9) hidden = agg2 * rsqrt(deg_in) + b2  -> d_out (second tuple element)
  {
    int n2 = N_NODES * OUT_F;
    epilogue_hidden<<<(n2 + 255) / 256, 256, 0, stream>>>(agg2, rinv_in, b2, hidden, n2);
  }
  // 10) logits = hidden @ Wf + bf  -> d_out (first tuple element)
  logits_kernel<<<(N_NODES + 255) / 256, 256, 0, stream>>>(hidden, Wf, bf, logits, N_NODES);
}